// TokenEmbedding_43276090474610
// MI455X (gfx1250) — compile-verified
//
#include <hip/hip_runtime.h>
#include <hip/hip_bf16.h>

// CDNA5 / gfx1250 segment-sum via V_WMMA_F32_16X16X4_F32 with a one-hot
// selection A-matrix. Sorted indices => each 16-token tile owns a contiguous
// wordpiece run => exclusive tile ownership, no atomics, no zero-init pass.
//
// Round-3 fix: seg loads made *unconditionally live* by folding the tail
// bounds check into the comparison target (want = in-run ? t0+m : -1) instead
// of the predicate, so the compiler can no longer sink a load under an
// exec-masked branch. Loop body is fully branchless; EXEC stays all-ones.

typedef float v2f __attribute__((ext_vector_type(2)));
typedef float v8f __attribute__((ext_vector_type(8)));

constexpr int B_ = 8;
constexpr int L_ = 4096;
constexpr int H_ = 1024;
constexpr int TILE_T = 16;        // token tile (WMMA M)
constexpr int NT = L_ / TILE_T;   // 256 token tiles per batch row
constexpr int NH = 8;             // 16-wide h-tiles per wave (WMMA N blocks)

// ---- Pre-pass: bounds[b * (NT+1) + i] = lower_bound(seg[b,:], i*16) ----
__global__ void seg16_bounds_kernel(const int* __restrict__ seg,
                                    int* __restrict__ bounds) {
  const int idx = blockIdx.x * blockDim.x + threadIdx.x;
  if (idx >= B_ * (NT + 1)) return;
  const int b = idx / (NT + 1);
  const int i = idx - b * (NT + 1);
  const int target = i * TILE_T;
  const int* __restrict__ srow = seg + (size_t)b * L_;
  int lo = 0, hi = L_;
  while (lo < hi) {
    int mid = (lo + hi) >> 1;
    if (srow[mid] < target) lo = mid + 1; else hi = mid;
  }
  bounds[idx] = lo;
}

// ---- Main kernel: one wave per (token tile, batch, 128-wide h slab) ----
__launch_bounds__(32)
__global__ void seg16_wmma_kernel(const float* __restrict__ X,
                                  const int* __restrict__ seg,
                                  const int* __restrict__ bounds,
                                  float* __restrict__ out) {
  const int tile = blockIdx.x;               // which 16-token output tile
  const int b    = blockIdx.y;               // batch row
  const int hb   = blockIdx.z * (NH * 16);   // h base for this wave
  const int t0   = tile * TILE_T;
  const int lane = threadIdx.x;              // 0..31 (wave32)
  const int mn   = lane & 15;                // M for A, N for B/C/D
  const int kb   = (lane >> 4) * 2;          // K pair handled by this half-wave

  const int* __restrict__ srow = seg + (size_t)b * L_;
  const int* __restrict__ brow = bounds + (size_t)b * (NT + 1);
  const int w_start = brow[tile];
  const int w_end   = brow[tile + 1];

  v8f C[NH];
#pragma unroll
  for (int t = 0; t < NH; ++t) C[t] = (v8f){0.f, 0.f, 0.f, 0.f, 0.f, 0.f, 0.f, 0.f};

  const float* __restrict__ Xb = X + (size_t)b * L_ * H_ + hb;

  // ---- accumulate run in K-steps of 4 wordpieces ----
  for (int w = w_start; w < w_end; w += 4) {
    const int k0 = w + kb;
    const int k1 = k0 + 1;
    // Clamp addresses to the row (memory safety only).
    const int r0 = (k0 < L_) ? k0 : (L_ - 1);
    const int r1 = (k1 < L_) ? k1 : (L_ - 1);

    // Unconditional seg loads; result is ALWAYS consumed by the compare below,
    // so the compiler cannot sink these under an exec-masked branch.
    const int s0 = srow[r0];
    const int s1 = srow[r1];

    // Tail kill via the comparison target: seg values are >= 0, so -1 never
    // matches. (For k in [w_end, L), srow[k] >= t0+16 can't match either;
    // the -1 also covers the w_end==L clamp corner.)
    const int want0 = (k0 < w_end) ? (t0 + mn) : -1;
    const int want1 = (k1 < w_end) ? (t0 + mn) : -1;

    // One-hot A (16x4 f32).
    v2f A;
    A.x = (s0 == want0) ? 1.0f : 0.0f;
    A.y = (s1 == want1) ? 1.0f : 0.0f;

    const float* __restrict__ x0 = Xb + (size_t)r0 * H_ + mn;
    const float* __restrict__ x1 = Xb + (size_t)r1 * H_ + mn;

    // Branchless speculative prefetch of the next K-step's rows.
    int pw = w + 4 + kb;
    pw = (pw < L_) ? pw : (L_ - 1);
    __builtin_prefetch(Xb + (size_t)pw * H_ + mn, 0, 1);

#pragma unroll
    for (int t = 0; t < NH; ++t) {
      // B (4x16 f32): rows kb/kb+1 at column n; 16-lane-contiguous => coalesced.
      v2f Bv;
      Bv.x = x0[t * 16];
      Bv.y = x1[t * 16];
      // D = A x B + C   (v_wmma_f32_16x16x4_f32)
      C[t] = __builtin_amdgcn_wmma_f32_16x16x4_f32(
          /*neg_a=*/false, A, /*neg_b=*/false, Bv,
          /*c_mod=*/(short)0, C[t], /*reuse_a=*/false, /*reuse_b=*/false);
    }
  }

  // ---- store D tiles (exclusive ownership: plain stores; empty run => zeros)
  // C/D layout: VGPR v -> row t0+v (lanes 0-15) / t0+8+v (lanes 16-31),
  // lane&15 -> column.
  float* __restrict__ orow =
      out + ((size_t)b * L_ + t0 + (lane >> 4) * 8) * H_ + hb + mn;
#pragma unroll
  for (int t = 0; t < NH; ++t) {
#pragma unroll
    for (int v = 0; v < 8; ++v) {
      orow[(size_t)v * H_ + t * 16] = C[t][v];
    }
  }
}

extern "C" void kernel_launch(void* const* d_in, const int* in_sizes, int n_in,
                              void* d_out, int out_size, void* d_ws, size_t ws_size,
                              hipStream_t stream) {
  const float* X  = (const float*)d_in[0];   // sequence_output  [B, L, H] f32
  const int* seg  = (const int*)d_in[1];     // wordpiece_to_token [B, L] i32
  float* out      = (float*)d_out;           // [B, L, H] f32
  int* bounds     = (int*)d_ws;              // B * (NT+1) ints (~8.2 KB)

  // Pre-pass: tile bounds via binary search (2056 threads total).
  const int nb = B_ * (NT + 1);
  hipLaunchKernelGGL(seg16_bounds_kernel, dim3((nb + 255) / 256), dim3(256),
                     0, stream, seg, bounds);

  // Main pass: (256, 8, 8) one-wave workgroups.
  dim3 grid(NT, B_, H_ / (NH * 16));
  hipLaunchKernelGGL(seg16_wmma_kernel, grid, dim3(32), 0, stream,
                     X, seg, bounds, out);
}